// HNO_50551765073969
// MI455X (gfx1250) — compile-verified
//
#include <hip/hip_runtime.h>

typedef __attribute__((ext_vector_type(2))) float v2f;
typedef __attribute__((ext_vector_type(8))) float v8f;

#define HDIM 64
#define BN_EPS 1e-5f

// ---------------- utility kernels ----------------

__global__ void zero_kernel(float* __restrict__ p, long n) {
  long i = (long)blockIdx.x * blockDim.x + threadIdx.x;
  long st = (long)gridDim.x * blockDim.x;
  for (; i < n; i += st) p[i] = 0.0f;
}

// pad x1 [N,14] -> out [N,16] with zero columns 14,15
__global__ void pad14to16_kernel(const float* __restrict__ x, float* __restrict__ out, long N) {
  long total = N * 16;
  long i = (long)blockIdx.x * blockDim.x + threadIdx.x;
  long st = (long)gridDim.x * blockDim.x;
  for (; i < total; i += st) {
    long r = i >> 4;
    int  c = (int)(i & 15);
    out[i] = (c < 14) ? x[r * 14 + c] : 0.0f;
  }
}

__global__ void degree_kernel(const int* __restrict__ dst, float* __restrict__ cnt, long E) {
  long i = (long)blockIdx.x * blockDim.x + threadIdx.x;
  long st = (long)gridDim.x * blockDim.x;
  for (; i < E; i += st) atomicAdd(&cnt[dst[i]], 1.0f);
}

// scatter-add for layer1: x stride 14 (original features), s stride 16 (padded sums)
__global__ void scatter_add14_kernel(const float* __restrict__ x, const int* __restrict__ src,
                                     const int* __restrict__ dst, float* __restrict__ s,
                                     long E) {
  long total = E * 14;
  long i = (long)blockIdx.x * blockDim.x + threadIdx.x;
  long st = (long)gridDim.x * blockDim.x;
  for (; i < total; i += st) {
    long e = i / 14;
    int  c = (int)(i - e * 14);
    atomicAdd(&s[(long)dst[e] * 16 + c], x[(long)src[e] * 14 + c]);
  }
}

// vectorized scatter-add for d==64: one float4 chunk per work-item
__global__ void scatter_add64(const float* __restrict__ x, const int* __restrict__ src,
                              const int* __restrict__ dst, float* __restrict__ s, long E) {
  long total = E * 16;  // 16 float4 chunks per edge
  long i = (long)blockIdx.x * blockDim.x + threadIdx.x;
  long st = (long)gridDim.x * blockDim.x;
  for (; i < total; i += st) {
    long e = i >> 4;
    int  q = (int)(i & 15);
    const float4 v = ((const float4*)(x + (long)src[e] * HDIM))[q];
    float* o = s + (long)dst[e] * HDIM + q * 4;
    atomicAdd(o + 0, v.x);
    atomicAdd(o + 1, v.y);
    atomicAdd(o + 2, v.z);
    atomicAdd(o + 3, v.w);
  }
}

// ---------------- fused SAGE layer: out = act(mean@Wn + x@Wr + b) ----------------
// One wave computes a 16x64 f32 output tile via v_wmma_f32_16x16x4_f32.
// A-frag (16x4 f32): lanes 0-15 -> M=lane, K={kb,kb+1}; lanes 16-31 -> M=lane-16, K={kb+2,kb+3}.
// Weights staged in LDS with {k,k+1} pairs interleaved per column -> one ds_load_b64 per B-frag.
// DIN is compile-time (16 or 64): no divergent guards anywhere in the math loop.
template <int DIN>
__global__ __launch_bounds__(256) void sage_wmma_kernel(
    const float* __restrict__ X, const float* __restrict__ S, const float* __restrict__ cnt,
    const float* __restrict__ Wn, const float* __restrict__ Wr, const float* __restrict__ bias,
    float* __restrict__ out, int N, int wrows /*valid rows in Wn/Wr (14 or 64)*/, int act) {
  __shared__ float lWn[DIN * HDIM];
  __shared__ float lWr[DIN * HDIM];
  // interleaved fill: element (k,c) -> lds[(k>>1)*128 + c*2 + (k&1)]
  for (int i = threadIdx.x; i < DIN * HDIM; i += 256) {
    int k = i >> 6;
    int c = i & 63;
    int idx = (k >> 1) * 128 + c * 2 + (k & 1);
    lWn[idx] = (k < wrows) ? Wn[k * HDIM + c] : 0.0f;
    lWr[idx] = (k < wrows) ? Wr[k * HDIM + c] : 0.0f;
  }
  __syncthreads();

  const int lane = threadIdx.x & 31;
  const int wv   = threadIdx.x >> 5;
  const int half = lane >> 4;
  const int mr   = lane & 15;
  const long tile = ((long)blockIdx.x * 8 + wv) * 16;
  if (tile >= N) return;  // wave-uniform: EXEC stays all-1s for active waves
  const long row = tile + mr;

  const float inv = 1.0f / fmaxf(cnt[row], 1.0f);
  const float* __restrict__ As = S + row * (long)DIN;
  const float* __restrict__ Ax = X + row * (long)DIN;

  v8f acc[4] = {};

#pragma unroll
  for (int pass = 0; pass < 2; ++pass) {
    const float* __restrict__ Arow = pass ? Ax : As;
    const float* __restrict__ W    = pass ? lWr : lWn;
    const float  scale = pass ? 1.0f : inv;
#pragma unroll
    for (int k0 = 0; k0 < DIN; k0 += 4) {
      const int kb = k0 + half * 2;  // even; 8B-aligned since row*DIN is even
      v2f a = *(const v2f*)(Arow + kb);
      a.x *= scale;
      a.y *= scale;
      const float* __restrict__ Wp = W + (kb >> 1) * 128;
#pragma unroll
      for (int j = 0; j < 4; ++j) {
        const int col = j * 16 + mr;
        v2f b = *(const v2f*)(Wp + col * 2);  // {W[kb][col], W[kb+1][col]}
        acc[j] = __builtin_amdgcn_wmma_f32_16x16x4_f32(false, a, false, b, (short)0,
                                                       acc[j], false, false);
      }
    }
  }

  // C/D layout: VGPR i -> (lanes 0-15: M=i), (lanes 16-31: M=8+i)
#pragma unroll
  for (int j = 0; j < 4; ++j) {
    const int col = j * 16 + mr;
    const float bv = bias[col];
#pragma unroll
    for (int i = 0; i < 8; ++i) {
      float z = acc[j][i] + bv;
      if (act == 1) z = (z > 0.0f) ? z : 0.01f * z;  // LeakyReLU(0.01)
      else if (act == 2) z = fmaxf(z, 0.0f);         // ReLU
      out[(tile + half * 8 + i) * HDIM + col] = z;
    }
  }
}

// ---------------- plain GEMM (MLP head): out[N,dout] = X[N,64] @ W[64,dout] (+bias) ----------------
// DOUTP: padded output width (64 or 32). Weight zero-padded into LDS; only the store masks cols.
template <int DOUTP>
__global__ __launch_bounds__(256) void gemm_wmma_kernel(
    const float* __restrict__ X, const float* __restrict__ W, const float* __restrict__ bias,
    float* __restrict__ out, int N, int dout) {
  __shared__ float lW[HDIM * DOUTP];
  for (int i = threadIdx.x; i < HDIM * DOUTP; i += 256) {
    int k = i / DOUTP;
    int c = i - k * DOUTP;
    int idx = (k >> 1) * (2 * DOUTP) + c * 2 + (k & 1);
    lW[idx] = (c < dout) ? W[k * dout + c] : 0.0f;
  }
  __syncthreads();

  const int lane = threadIdx.x & 31;
  const int wv   = threadIdx.x >> 5;
  const int half = lane >> 4;
  const int mr   = lane & 15;
  const long tile = ((long)blockIdx.x * 8 + wv) * 16;
  if (tile >= N) return;
  const long row = tile + mr;

  constexpr int NT = DOUTP / 16;
  const float* __restrict__ Arow = X + row * HDIM;

  v8f acc[NT] = {};

#pragma unroll
  for (int k0 = 0; k0 < HDIM; k0 += 4) {
    const int kb = k0 + half * 2;
    v2f a = *(const v2f*)(Arow + kb);
    const float* __restrict__ Wp = lW + (kb >> 1) * (2 * DOUTP);
#pragma unroll
    for (int j = 0; j < NT; ++j) {
      const int col = j * 16 + mr;
      v2f b = *(const v2f*)(Wp + col * 2);
      acc[j] = __builtin_amdgcn_wmma_f32_16x16x4_f32(false, a, false, b, (short)0,
                                                     acc[j], false, false);
    }
  }

#pragma unroll
  for (int j = 0; j < NT; ++j) {
    const int col = j * 16 + mr;
    if (col < dout) {
      const float bv = bias ? bias[col] : 0.0f;
#pragma unroll
      for (int i = 0; i < 8; ++i) {
        out[(tile + half * 8 + i) * (long)dout + col] = acc[j][i] + bv;
      }
    }
  }
}

// ---------------- BatchNorm (batch stats, biased var) ----------------

__global__ __launch_bounds__(256) void colstats_kernel(const float* __restrict__ x,
                                                       float* __restrict__ stats, int N) {
  const int c  = threadIdx.x & 63;
  const int rg = threadIdx.x >> 6;  // 4 row-groups per block
  float s = 0.0f, s2 = 0.0f;
  for (long r = (long)blockIdx.x * 4 + rg; r < N; r += (long)gridDim.x * 4) {
    float v = x[r * HDIM + c];
    s += v;
    s2 += v * v;
  }
  __shared__ float red[256], red2[256];
  red[threadIdx.x]  = s;
  red2[threadIdx.x] = s2;
  __syncthreads();
  if (rg == 0) {
    s  = red[c]  + red[64 + c]  + red[128 + c]  + red[192 + c];
    s2 = red2[c] + red2[64 + c] + red2[128 + c] + red2[192 + c];
    atomicAdd(&stats[c], s);
    atomicAdd(&stats[64 + c], s2);
  }
}

__global__ void bn_apply_kernel(float* __restrict__ x, const float* __restrict__ stats,
                                const float* __restrict__ g, const float* __restrict__ be,
                                int N, int relu) {
  long total = (long)N * HDIM;
  long i = (long)blockIdx.x * blockDim.x + threadIdx.x;
  long st = (long)gridDim.x * blockDim.x;
  const float invN = 1.0f / (float)N;
  for (; i < total; i += st) {
    int c = (int)(i & 63);
    float m = stats[c] * invN;
    float v = stats[64 + c] * invN - m * m;
    float z = (x[i] - m) * rsqrtf(v + BN_EPS) * g[c] + be[c];
    if (relu) z = fmaxf(z, 0.0f);
    x[i] = z;
  }
}

// ---------------- host orchestration ----------------

extern "C" void kernel_launch(void* const* d_in, const int* in_sizes, int n_in,
                              void* d_out, int out_size, void* d_ws, size_t ws_size,
                              hipStream_t stream) {
  const int  N = in_sizes[0] / 14;
  const long E = in_sizes[1] / 2;
  const float* x1  = (const float*)d_in[0];
  const int*   src = (const int*)d_in[1];
  const int*   dst = src + E;

  const float* Wn1 = (const float*)d_in[2];  const float* Wr1 = (const float*)d_in[3];  const float* b1 = (const float*)d_in[4];
  const float* Wn2 = (const float*)d_in[5];  const float* Wr2 = (const float*)d_in[6];  const float* b2 = (const float*)d_in[7];
  const float* Wn3 = (const float*)d_in[8];  const float* Wr3 = (const float*)d_in[9];  const float* b3 = (const float*)d_in[10];
  const float* Wn4 = (const float*)d_in[11]; const float* Wr4 = (const float*)d_in[12]; const float* b4 = (const float*)d_in[13];
  const float* g1  = (const float*)d_in[14]; const float* be1 = (const float*)d_in[15];
  const float* g2  = (const float*)d_in[16]; const float* be2 = (const float*)d_in[17];
  const float* g3  = (const float*)d_in[18]; const float* be3 = (const float*)d_in[19];
  const float* Wm0 = (const float*)d_in[20]; const float* gm0 = (const float*)d_in[21]; const float* bem0 = (const float*)d_in[22];
  const float* Wm1 = (const float*)d_in[23]; const float* gm1 = (const float*)d_in[24]; const float* bem1 = (const float*)d_in[25];
  const float* Wm2 = (const float*)d_in[26]; const float* bm2 = (const float*)d_in[27];

  float* ws    = (float*)d_ws;
  float* cnt   = ws;  ws += N;
  float* S     = ws;  ws += (long)N * HDIM;
  float* A     = ws;  ws += (long)N * HDIM;
  float* B     = ws;  ws += (long)N * HDIM;
  float* stats = ws;  // 128 floats

  const int TB = 256;
  dim3 sgrid((unsigned)((N + 127) / 128));

  auto bn = [&](float* buf, const float* g, const float* be, int relu) {
    zero_kernel<<<1, 128, 0, stream>>>(stats, 128);
    colstats_kernel<<<512, TB, 0, stream>>>(buf, stats, N);
    bn_apply_kernel<<<2048, TB, 0, stream>>>(buf, stats, g, be, N, relu);
  };

  // degrees (shared by all layers)
  zero_kernel<<<512, TB, 0, stream>>>(cnt, N);
  degree_kernel<<<1024, TB, 0, stream>>>(dst, cnt, E);

  // conv1 (din=14, padded to 16) + LeakyReLU + BN
  pad14to16_kernel<<<2048, TB, 0, stream>>>(x1, B, N);          // B[:, :16] = padded x1
  zero_kernel<<<2048, TB, 0, stream>>>(S, (long)N * 16);        // S[:, :16] = neighbor sums
  scatter_add14_kernel<<<4096, TB, 0, stream>>>(x1, src, dst, S, E);
  sage_wmma_kernel<16><<<sgrid, TB, 0, stream>>>(B, S, cnt, Wn1, Wr1, b1, A, N, 14, 1);
  bn(A, g1, be1, 0);

  // conv2 + LeakyReLU + BN
  zero_kernel<<<2048, TB, 0, stream>>>(S, (long)N * HDIM);
  scatter_add64<<<4096, TB, 0, stream>>>(A, src, dst, S, E);
  sage_wmma_kernel<HDIM><<<sgrid, TB, 0, stream>>>(A, S, cnt, Wn2, Wr2, b2, B, N, HDIM, 1);
  bn(B, g2, be2, 0);

  // conv3 + ReLU + BN
  zero_kernel<<<2048, TB, 0, stream>>>(S, (long)N * HDIM);
  scatter_add64<<<4096, TB, 0, stream>>>(B, src, dst, S, E);
  sage_wmma_kernel<HDIM><<<sgrid, TB, 0, stream>>>(B, S, cnt, Wn3, Wr3, b3, A, N, HDIM, 2);
  bn(A, g3, be3, 0);

  // conv4 (no act, no BN)
  zero_kernel<<<2048, TB, 0, stream>>>(S, (long)N * HDIM);
  scatter_add64<<<4096, TB, 0, stream>>>(A, src, dst, S, E);
  sage_wmma_kernel<HDIM><<<sgrid, TB, 0, stream>>>(A, S, cnt, Wn4, Wr4, b4, B, N, HDIM, 0);

  // MLP head: Linear->BN->ReLU, Linear->BN->ReLU, Linear+bias
  gemm_wmma_kernel<64><<<sgrid, TB, 0, stream>>>(B, Wm0, nullptr, A, N, HDIM);
  bn(A, gm0, bem0, 1);
  gemm_wmma_kernel<64><<<sgrid, TB, 0, stream>>>(A, Wm1, nullptr, B, N, HDIM);
  bn(B, gm1, bem1, 1);
  gemm_wmma_kernel<32><<<sgrid, TB, 0, stream>>>(B, Wm2, bm2, (float*)d_out, N, 21);
}